// GIN_node_classification_54245436949040
// MI455X (gfx1250) — compile-verified
//
#include <hip/hip_runtime.h>

typedef __attribute__((ext_vector_type(2))) float v2f;
typedef __attribute__((ext_vector_type(8))) float v8f;

#define BN_EPS 1e-5f

// ---------------------------------------------------------------------------
// 1) BN0 (eval mode) + initialize agg1 = h0 (GIN self term). float4 vectorized.
// ---------------------------------------------------------------------------
__global__ void bn0_init_kernel(const float4* __restrict__ x,
                                const float4* __restrict__ gamma,
                                const float4* __restrict__ beta,
                                const float4* __restrict__ mean,
                                const float4* __restrict__ var,
                                float4* __restrict__ h0,
                                float4* __restrict__ agg1,
                                int total4) {
    int i = blockIdx.x * blockDim.x + threadIdx.x;
    if (i >= total4) return;
    int f4 = i & 31;  // F_IN/4 = 32 float4 groups per row
    float4 g = gamma[f4], be = beta[f4], mu = mean[f4], va = var[f4];
    float4 xv = x[i];
    float4 o;
    o.x = (xv.x - mu.x) * (g.x * rsqrtf(va.x + BN_EPS)) + be.x;
    o.y = (xv.y - mu.y) * (g.y * rsqrtf(va.y + BN_EPS)) + be.y;
    o.z = (xv.z - mu.z) * (g.z * rsqrtf(va.z + BN_EPS)) + be.z;
    o.w = (xv.w - mu.w) * (g.w * rsqrtf(va.w + BN_EPS)) + be.w;
    h0[i] = o;
    agg1[i] = o;
}

// ---------------------------------------------------------------------------
// 2/4) Edge scatter-add: one edge per wave32; b128 gather + 4 atomics/lane/iter.
//      Feature tables (25.6/51.2 MB) are resident in the 192MB L2, so both the
//      random-row gathers and the f32 atomics stay on-die.
// ---------------------------------------------------------------------------
template <int F>
__global__ void scatter_add_kernel(const float* __restrict__ h,
                                   float* __restrict__ agg,
                                   const int* __restrict__ src,
                                   const int* __restrict__ dst,
                                   int E) {
    int wave = (int)((blockIdx.x * blockDim.x + threadIdx.x) >> 5);
    int lane = threadIdx.x & 31;
    if (wave >= E) return;
    int s = src[wave];
    int d = dst[wave];
    const float4* hs = (const float4*)(h + (long long)s * F);
    float* ad = agg + (long long)d * F;
#pragma unroll
    for (int i = 0; i < F / 128; ++i) {
        float4 v = hs[lane + i * 32];
        int f = (lane + i * 32) * 4;
        atomicAdd(&ad[f + 0], v.x);
        atomicAdd(&ad[f + 1], v.y);
        atomicAdd(&ad[f + 2], v.z);
        atomicAdd(&ad[f + 3], v.w);
    }
}

// ---------------------------------------------------------------------------
// 3) GEMM1: h1 = relu(BN1(agg1[N,128] @ W1[128,256] + b1)); also agg2 = h1.
//    One wave per block; 32 rows x 64 cols per wave = 8 WMMA accumulators.
//    B fragments reused across both row tiles. Row/col indices are CLAMPED
//    (not predicated): garbage only reaches output rows masked at store,
//    so the hot loop has zero exec-mask manipulation.
// ---------------------------------------------------------------------------
__global__ __launch_bounds__(32) void gemm1_bn_relu_kernel(
    const float* __restrict__ A,     // agg1 [N,128]
    const float* __restrict__ W,     // W1   [128,256] row-major
    const float* __restrict__ bias,  // b1   [256]
    const float* __restrict__ g1, const float* __restrict__ be1,
    const float* __restrict__ mu1, const float* __restrict__ va1,
    float* __restrict__ h1, float* __restrict__ agg2, int N) {
    const int lane = threadIdx.x & 31;
    const int m    = lane & 15;
    const int half = lane >> 4;      // selects K pair within the 16x4 fragment
    const int row0 = blockIdx.x * 32;
    const int n0   = blockIdx.y * 64;

    v8f acc[2][4] = {};              // [row tile][col tile]

    const int rA0 = row0 + m;
    const int rA1 = row0 + 16 + m;
    const float* A0 = A + (long long)(rA0 < N ? rA0 : N - 1) * 128;
    const float* A1 = A + (long long)(rA1 < N ? rA1 : N - 1) * 128;

    for (int k0 = 0; k0 < 128; k0 += 4) {
        const int ka = k0 + half * 2;
        v2f a0 = *(const v2f*)(A0 + ka);   // A[m, ka..ka+1] (8B aligned)
        v2f a1 = *(const v2f*)(A1 + ka);
#pragma unroll
        for (int t = 0; t < 4; ++t) {
            const float* Wk = W + (long long)ka * 256 + (n0 + t * 16 + m);
            v2f b;
            b.x = Wk[0];     // B[ka,   col]
            b.y = Wk[256];   // B[ka+1, col]
            acc[0][t] = __builtin_amdgcn_wmma_f32_16x16x4_f32(
                false, a0, false, b, (short)0, acc[0][t], false, false);
            acc[1][t] = __builtin_amdgcn_wmma_f32_16x16x4_f32(
                false, a1, false, b, (short)0, acc[1][t], false, false);
        }
    }

    // Fused epilogue: bias + BN1(eval) + ReLU; write h1 and init agg2.
#pragma unroll
    for (int t = 0; t < 4; ++t) {
        const int c  = n0 + t * 16 + m;
        const float sc = g1[c] * rsqrtf(va1[c] + BN_EPS);
        const float mu = mu1[c], bb = be1[c], bs = bias[c];
#pragma unroll
        for (int rt = 0; rt < 2; ++rt) {
#pragma unroll
            for (int r = 0; r < 8; ++r) {
                const int row = row0 + rt * 16 + r + 8 * half;
                if (row < N) {
                    float v = acc[rt][t][r] + bs;
                    v = (v - mu) * sc + bb;
                    v = fmaxf(v, 0.0f);
                    const long long idx = (long long)row * 256 + c;
                    h1[idx]   = v;
                    agg2[idx] = v;
                }
            }
        }
    }
}

// ---------------------------------------------------------------------------
// 5) GEMM2: out = agg2[N,256] @ W2[256,40] + b2. NCLS=40 -> 3 col-tiles of 16.
//    32 rows per wave (B reuse); clamped loads, guarded stores only.
// ---------------------------------------------------------------------------
__global__ __launch_bounds__(32) void gemm2_kernel(
    const float* __restrict__ A,     // agg2 [N,256]
    const float* __restrict__ W,     // W2   [256,40] row-major
    const float* __restrict__ bias,  // b2   [40]
    float* __restrict__ out, int N) {
    const int lane = threadIdx.x & 31;
    const int m    = lane & 15;
    const int half = lane >> 4;
    const int row0 = blockIdx.x * 32;
    const int n0   = blockIdx.y * 16;
    const int col  = n0 + m;
    const int cc   = col < 40 ? col : 39;   // clamped B column

    v8f acc[2] = {};
    const int rA0 = row0 + m;
    const int rA1 = row0 + 16 + m;
    const float* A0 = A + (long long)(rA0 < N ? rA0 : N - 1) * 256;
    const float* A1 = A + (long long)(rA1 < N ? rA1 : N - 1) * 256;

    for (int k0 = 0; k0 < 256; k0 += 4) {
        const int ka = k0 + half * 2;
        v2f a0 = *(const v2f*)(A0 + ka);
        v2f a1 = *(const v2f*)(A1 + ka);
        v2f b;
        b.x = W[(long long)ka * 40 + cc];
        b.y = W[(long long)(ka + 1) * 40 + cc];
        acc[0] = __builtin_amdgcn_wmma_f32_16x16x4_f32(
            false, a0, false, b, (short)0, acc[0], false, false);
        acc[1] = __builtin_amdgcn_wmma_f32_16x16x4_f32(
            false, a1, false, b, (short)0, acc[1], false, false);
    }

    const float bs = bias[cc];
#pragma unroll
    for (int rt = 0; rt < 2; ++rt) {
#pragma unroll
        for (int r = 0; r < 8; ++r) {
            const int row = row0 + rt * 16 + r + 8 * half;
            if (row < N && col < 40) {
                out[(long long)row * 40 + col] = acc[rt][r] + bs;
            }
        }
    }
}

// ---------------------------------------------------------------------------
// Host launcher
// Inputs (setup_inputs order): x, edge_index, bn0_{gamma,beta,mean,var},
//                              W1, b1, bn1_{gamma,beta,mean,var}, W2, b2
// ---------------------------------------------------------------------------
extern "C" void kernel_launch(void* const* d_in, const int* in_sizes, int n_in,
                              void* d_out, int out_size, void* d_ws, size_t ws_size,
                              hipStream_t stream) {
    const float* x     = (const float*)d_in[0];
    const int*   edge  = (const int*)d_in[1];   // [2, E] (src row, then dst row)
    const float* bn0_g = (const float*)d_in[2];
    const float* bn0_b = (const float*)d_in[3];
    const float* bn0_m = (const float*)d_in[4];
    const float* bn0_v = (const float*)d_in[5];
    const float* W1    = (const float*)d_in[6];
    const float* b1    = (const float*)d_in[7];
    const float* bn1_g = (const float*)d_in[8];
    const float* bn1_b = (const float*)d_in[9];
    const float* bn1_m = (const float*)d_in[10];
    const float* bn1_v = (const float*)d_in[11];
    const float* W2    = (const float*)d_in[12];
    const float* b2    = (const float*)d_in[13];
    float* out = (float*)d_out;

    const int FIN = 128, HID = 256;
    const int N = in_sizes[0] / FIN;
    const int E = in_sizes[1] / 2;
    const int* src = edge;
    const int* dst = edge + E;

    // Workspace layout (fp32): h0[N*128] | agg1[N*128] | h1[N*256] | agg2[N*256]
    float* h0   = (float*)d_ws;
    float* agg1 = h0   + (size_t)N * FIN;
    float* h1   = agg1 + (size_t)N * FIN;
    float* agg2 = h1   + (size_t)N * HID;

    // 1) BN0 + agg1 init (float4 vectorized)
    {
        const int total4 = N * FIN / 4;
        bn0_init_kernel<<<(total4 + 255) / 256, 256, 0, stream>>>(
            (const float4*)x, (const float4*)bn0_g, (const float4*)bn0_b,
            (const float4*)bn0_m, (const float4*)bn0_v,
            (float4*)h0, (float4*)agg1, total4);
    }
    // 2) scatter-add layer 1 (one edge per wave, 8 waves per block)
    {
        const int grid = (E + 7) / 8;
        scatter_add_kernel<128><<<grid, 256, 0, stream>>>(h0, agg1, src, dst, E);
    }
    // 3) GEMM1 + bias + BN1 + ReLU (+ agg2 init)
    {
        dim3 grid((N + 31) / 32, HID / 64);
        gemm1_bn_relu_kernel<<<grid, 32, 0, stream>>>(
            agg1, W1, b1, bn1_g, bn1_b, bn1_m, bn1_v, h1, agg2, N);
    }
    // 4) scatter-add layer 2
    {
        const int grid = (E + 7) / 8;
        scatter_add_kernel<256><<<grid, 256, 0, stream>>>(h1, agg2, src, dst, E);
    }
    // 5) GEMM2 + bias -> logits [N, 40]
    {
        dim3 grid((N + 31) / 32, 3);
        gemm2_kernel<<<grid, 32, 0, stream>>>(agg2, W2, b2, out, N);
    }
}